// APPNP_6124623364171
// MI455X (gfx1250) — compile-verified
//
#include <hip/hip_runtime.h>
#include <hip/hip_bf16.h>

typedef __attribute__((ext_vector_type(2))) float v2f;
typedef __attribute__((ext_vector_type(8))) float v8f;

#define IN_FEATS 512
#define HIDDEN   64
#define NCLASSES 47
#define ALPHA    0.1f
#define KSTEPS   10

// ---------------------------------------------------------------------------
// utility: zero a buffer (float4 granularity; counts divide evenly here)
// ---------------------------------------------------------------------------
__global__ __launch_bounds__(256) void appnp_zero4(float4* __restrict__ p, int n4) {
    int i = blockIdx.x * blockDim.x + threadIdx.x;
    if (i < n4) p[i] = make_float4(0.f, 0.f, 0.f, 0.f);
}

// ---------------------------------------------------------------------------
// degree accumulation: out_deg[src[e]] += 1, in_deg[dst[e]] += 1
// ---------------------------------------------------------------------------
__global__ __launch_bounds__(256) void appnp_degree(const int* __restrict__ src,
                                                    const int* __restrict__ dst,
                                                    float* __restrict__ odeg,
                                                    float* __restrict__ ideg, int E) {
    int e = blockIdx.x * blockDim.x + threadIdx.x;
    if (e >= E) return;
    atomicAdd(&odeg[src[e]], 1.0f);
    atomicAdd(&ideg[dst[e]], 1.0f);
}

// in-place: deg -> rsqrt(max(deg,1))  (both arrays are contiguous in ws)
__global__ __launch_bounds__(256) void appnp_norm(float* __restrict__ d, int n) {
    int i = blockIdx.x * blockDim.x + threadIdx.x;
    if (i < n) d[i] = rsqrtf(fmaxf(d[i], 1.0f));
}

// ---------------------------------------------------------------------------
// GEMM1: H = relu(F[N,512] @ W0[512,64] + b0) — fp32 WMMA 16x16x4.
// One wave owns a 16-row tile and all 4 col tiles (A fragment reused 4x); the
// 205MB feature matrix streams from HBM exactly once (the roofline limit).
// W0 (128KB) is staged through LDS in two 64KB chunks so B fragments come
// from ds_load instead of thrashing L0.  Out-of-range waves clamp their tile
// (duplicate work, identical stores) so block barriers stay uniform.
// ---------------------------------------------------------------------------
__global__ __launch_bounds__(256) void appnp_gemm1_wmma(const float* __restrict__ F,
                                                        const float* __restrict__ W0,
                                                        const float* __restrict__ b0,
                                                        float* __restrict__ H, int N) {
    __shared__ float sw[256 * HIDDEN];        // 64KB stage (of 320KB WGP LDS)
    const int lane = threadIdx.x & 31;
    const int wave = threadIdx.x >> 5;
    const int nTiles = (N + 15) >> 4;
    int rowTile = blockIdx.x * 8 + wave;
    if (rowTile >= nTiles) rowTile = nTiles - 1;   // clamp, keep barriers uniform
    const int rowBase = rowTile * 16;
    const int m  = lane & 15;                 // row (A) / col (B,C) within tile
    const int kh = (lane >> 4) << 1;          // K sub-offset: 0 or 2

    v8f acc0 = {}, acc1 = {}, acc2 = {}, acc3 = {};
    int arow = rowBase + m; if (arow >= N) arow = N - 1;
    const float* frow = F + (size_t)arow * IN_FEATS + kh;

    for (int ko = 0; ko < IN_FEATS; ko += 256) {
        __syncthreads();                      // previous chunk fully consumed
        const float4* gsrc = (const float4*)(W0 + (size_t)ko * HIDDEN);
        float4* ldst = (float4*)sw;
        for (int i = threadIdx.x; i < (256 * HIDDEN) / 4; i += 256) ldst[i] = gsrc[i];
        __syncthreads();

        for (int k = 0; k < 256; k += 4) {
            if ((k & 63) == 0) __builtin_prefetch(frow + ko + k + 64, 0, 1);
            v2f a = *(const v2f*)(frow + ko + k);      // A: 2 f32 per lane
            const float* w0p = sw + (k + kh) * HIDDEN + m;
            const float* w1p = w0p + HIDDEN;
            v2f b;
            b.x = w0p[0];  b.y = w1p[0];
            acc0 = __builtin_amdgcn_wmma_f32_16x16x4_f32(false, a, false, b, (short)0, acc0, false, false);
            b.x = w0p[16]; b.y = w1p[16];
            acc1 = __builtin_amdgcn_wmma_f32_16x16x4_f32(false, a, false, b, (short)0, acc1, false, false);
            b.x = w0p[32]; b.y = w1p[32];
            acc2 = __builtin_amdgcn_wmma_f32_16x16x4_f32(false, a, false, b, (short)0, acc2, false, false);
            b.x = w0p[48]; b.y = w1p[48];
            acc3 = __builtin_amdgcn_wmma_f32_16x16x4_f32(false, a, false, b, (short)0, acc3, false, false);
        }
    }

    // C/D layout: VGPR r, lanes 0-15 -> M=r ; lanes 16-31 -> M=8+r ; N = lane&15
    const int rh = (lane >> 4) * 8;
    const float bias0 = b0[m], bias1 = b0[16 + m], bias2 = b0[32 + m], bias3 = b0[48 + m];
    for (int r = 0; r < 8; ++r) {
        int row = rowBase + rh + r;
        if (row < N) {
            float* hrow = H + (size_t)row * HIDDEN;
            hrow[m]      = fmaxf(acc0[r] + bias0, 0.f);
            hrow[16 + m] = fmaxf(acc1[r] + bias1, 0.f);
            hrow[32 + m] = fmaxf(acc2[r] + bias2, 0.f);
            hrow[48 + m] = fmaxf(acc3[r] + bias3, 0.f);
        }
    }
}

// ---------------------------------------------------------------------------
// GEMM2: O = H[N,64] @ W1[64,47] + b1 — fp32 WMMA, 3 col tiles (last masked).
// Emits both h0 (teleport term) and hs = h0 * snorm (pre-scaled for scatter).
// ---------------------------------------------------------------------------
__global__ __launch_bounds__(256) void appnp_gemm2_wmma(const float* __restrict__ H,
                                                        const float* __restrict__ W1,
                                                        const float* __restrict__ b1,
                                                        const float* __restrict__ snorm,
                                                        float* __restrict__ O,
                                                        float* __restrict__ OS, int N) {
    const int lane = threadIdx.x & 31;
    const int wave = threadIdx.x >> 5;
    const int rowBase = (blockIdx.x * 8 + wave) * 16;
    if (rowBase >= N) return;                 // wave-uniform, no barriers here
    const int m  = lane & 15;
    const int kh = (lane >> 4) << 1;
    const bool c2ok = (32 + m) < NCLASSES;

    v8f acc0 = {}, acc1 = {}, acc2 = {};
    const float* hrow = H + (size_t)(rowBase + m) * HIDDEN + kh;

    for (int k = 0; k < HIDDEN; k += 4) {
        v2f a = *(const v2f*)(hrow + k);
        const float* w0p = W1 + (size_t)(k + kh) * NCLASSES + m;
        const float* w1p = w0p + NCLASSES;
        v2f b;
        b.x = w0p[0];  b.y = w1p[0];
        acc0 = __builtin_amdgcn_wmma_f32_16x16x4_f32(false, a, false, b, (short)0, acc0, false, false);
        b.x = w0p[16]; b.y = w1p[16];
        acc1 = __builtin_amdgcn_wmma_f32_16x16x4_f32(false, a, false, b, (short)0, acc1, false, false);
        b.x = c2ok ? w0p[32] : 0.f;
        b.y = c2ok ? w1p[32] : 0.f;
        acc2 = __builtin_amdgcn_wmma_f32_16x16x4_f32(false, a, false, b, (short)0, acc2, false, false);
    }

    const int rh = (lane >> 4) * 8;
    const float bias0 = b1[m], bias1 = b1[16 + m], bias2 = c2ok ? b1[32 + m] : 0.f;
    for (int r = 0; r < 8; ++r) {
        int row = rowBase + rh + r;
        float sn = snorm[row];
        float* orow = O  + (size_t)row * NCLASSES;
        float* srow = OS + (size_t)row * NCLASSES;
        float v0 = acc0[r] + bias0;
        float v1 = acc1[r] + bias1;
        orow[m]      = v0;  srow[m]      = v0 * sn;
        orow[16 + m] = v1;  srow[16 + m] = v1 * sn;
        if (c2ok) {
            float v2 = acc2[r] + bias2;
            orow[32 + m] = v2;  srow[32 + m] = v2 * sn;
        }
    }
}

// ---------------------------------------------------------------------------
// Edge scatter: acc[dst[e], c] += hs[src[e], c]   (hs is pre-scaled by snorm)
// Flat (edge,feature) indexing -> contiguous 47-float runs; the 18.8MB node
// matrix is resident in the 192MB L2, so gathers and atomics hit L2.
// ---------------------------------------------------------------------------
__global__ __launch_bounds__(256) void appnp_scatter(const float* __restrict__ hs,
                                                     const int* __restrict__ src,
                                                     const int* __restrict__ dst,
                                                     float* __restrict__ acc,
                                                     long long total) {
    long long idx = (long long)blockIdx.x * blockDim.x + threadIdx.x;
    if (idx >= total) return;
    int e = (int)(idx / NCLASSES);
    int c = (int)(idx - (long long)e * NCLASSES);
    atomicAdd(&acc[(size_t)dst[e] * NCLASSES + c],
              hs[(size_t)src[e] * NCLASSES + c]);
}

// mixed = (1-a)*acc*dnorm[row] + a*h0 ; out = do_scale ? mixed*snorm : mixed
__global__ __launch_bounds__(256) void appnp_mix(const float* __restrict__ acc,
                                                 const float* __restrict__ dnorm,
                                                 const float* __restrict__ snorm,
                                                 const float* __restrict__ h0,
                                                 float* __restrict__ out,
                                                 int do_scale, int total) {
    int idx = blockIdx.x * blockDim.x + threadIdx.x;
    if (idx >= total) return;
    int row = idx / NCLASSES;
    float mixed = (1.0f - ALPHA) * acc[idx] * dnorm[row] + ALPHA * h0[idx];
    out[idx] = do_scale ? mixed * snorm[row] : mixed;
}

// ---------------------------------------------------------------------------
extern "C" void kernel_launch(void* const* d_in, const int* in_sizes, int n_in,
                              void* d_out, int out_size, void* d_ws, size_t ws_size,
                              hipStream_t stream) {
    const float* F  = (const float*)d_in[0];
    const float* W0 = (const float*)d_in[1];
    const float* b0 = (const float*)d_in[2];
    const float* W1 = (const float*)d_in[3];
    const float* b1 = (const float*)d_in[4];
    const int*  src = (const int*)d_in[5];
    const int*  dst = (const int*)d_in[6];

    const int N = in_sizes[0] / IN_FEATS;   // 100000
    const int E = in_sizes[5];              // 1600000
    float* out = (float*)d_out;

    // workspace layout (floats); all offsets 16B-aligned for N=100000
    float* ws    = (float*)d_ws;
    float* snorm = ws;                           // [N]
    float* dnorm = ws + (size_t)N;               // [N]
    float* hrelu = ws + 2 * (size_t)N;           // [N,64]
    float* h0    = hrelu + (size_t)N * HIDDEN;   // [N,47] teleport term
    float* hs    = h0 + (size_t)N * NCLASSES;    // [N,47] snorm-scaled current
    float* acc   = hs + (size_t)N * NCLASSES;    // [N,47] scatter accumulator

    // 1) degrees + symmetric norms
    {
        int n4 = (2 * N) / 4;
        appnp_zero4<<<(n4 + 255) / 256, 256, 0, stream>>>((float4*)snorm, n4);
        appnp_degree<<<(E + 255) / 256, 256, 0, stream>>>(src, dst, snorm, dnorm, E);
        appnp_norm<<<(2 * N + 255) / 256, 256, 0, stream>>>(snorm, 2 * N);
    }

    // 2) MLP via fp32 WMMA (gemm2 also emits the pre-scaled hs)
    {
        int rowTiles = (N + 15) / 16;
        int blocks   = (rowTiles + 7) / 8;    // 8 waves/block, 1 row-tile/wave
        appnp_gemm1_wmma<<<blocks, 256, 0, stream>>>(F, W0, b0, hrelu, N);
        appnp_gemm2_wmma<<<blocks, 256, 0, stream>>>(hrelu, W1, b1, snorm, h0, hs, N);
    }

    // 3) K propagation steps; scatter reads pre-scaled hs, mix regenerates it
    //    (single accumulator buffer; last mix writes unscaled straight to d_out)
    {
        const long long tot = (long long)E * NCLASSES;
        const int sblocks = (int)((tot + 255) / 256);
        const int nc  = N * NCLASSES;
        const int nc4 = nc / 4;
        for (int it = 0; it < KSTEPS; ++it) {
            appnp_zero4<<<(nc4 + 255) / 256, 256, 0, stream>>>((float4*)acc, nc4);
            appnp_scatter<<<sblocks, 256, 0, stream>>>(hs, src, dst, acc, tot);
            if (it == KSTEPS - 1) {
                appnp_mix<<<(nc + 255) / 256, 256, 0, stream>>>(acc, dnorm, snorm, h0, out, 0, nc);
            } else {
                appnp_mix<<<(nc + 255) / 256, 256, 0, stream>>>(acc, dnorm, snorm, h0, hs, 1, nc);
            }
        }
    }
}